// GaborFilter_51462298140852
// MI455X (gfx1250) — compile-verified
//
#include <hip/hip_runtime.h>
#include <hip/hip_bf16.h>

typedef __attribute__((ext_vector_type(2))) float v2f;
typedef __attribute__((ext_vector_type(4))) float v4f;
typedef __attribute__((ext_vector_type(8))) float v8f;

#define BATCH 16
#define CHANS 32
#define ORI   8
#define HW    224
#define KS    7
#define YB    8              // output rows per block
#define NYB   (HW / YB)      // 28 row-bands
#define NXT   (HW / 16)      // 14 x-tiles per row
#define LDSROWS (YB + KS - 1) // 14
#define LDSW  232            // 224 + 6 halo, padded to 232

__global__ __launch_bounds__(256)
void GaborFilter_wmma_kernel(const float* __restrict__ x,
                             const float* __restrict__ filt,
                             float* __restrict__ out) {
  __shared__ float lds[LDSROWS * LDSW];

  const int tid  = threadIdx.x;
  const int lane = tid & 31;
  const int wave = tid >> 5;

  const int b   = blockIdx.x;
  const int n   = b / (CHANS * NYB);
  const int rem = b - n * (CHANS * NYB);
  const int c   = rem / NYB;
  const int yb  = rem - c * NYB;
  const int y0  = yb * YB;

  const float* __restrict__ xin = x + (size_t)(n * CHANS + c) * (HW * HW);

  // ---- Stage input band (14 rows x 232 cols, halo 3, zero-padded) into LDS
  for (int idx = tid; idx < LDSROWS * LDSW; idx += 256) {
    const int r  = idx / LDSW;
    const int cc = idx - r * LDSW;
    const int gy = y0 - 3 + r;
    const int gx = cc - 3;
    float v = 0.0f;
    if (gy >= 0 && gy < HW && gx >= 0 && gx < HW)
      v = xin[gy * HW + gx];
    lds[idx] = v;
  }

  // ---- B matrix (filters) into registers, once per block.
  // lane role: nidx = orientation column (and pixel row for A), half selects K-pair.
  const int nidx = lane & 15;
  const int half = lane >> 4;
  const float* __restrict__ fbase = filt + (size_t)(c * ORI + nidx) * (KS * KS);

  v2f bmat[14];
  #pragma unroll
  for (int g = 0; g < 14; ++g) {
    const int ky  = g >> 1;
    const int kx0 = (g & 1) * 4 + 2 * half;   // first of this lane's K-pair
    float b0 = 0.0f, b1 = 0.0f;
    if (nidx < ORI) {
      b0 = fbase[ky * KS + kx0];              // kx0 <= 6 always
      if (kx0 + 1 < KS) b1 = fbase[ky * KS + kx0 + 1];  // kx==7 is K-pad -> 0
    }
    v2f bm = {b0, b1};
    bmat[g] = bm;
  }

  __syncthreads();

  const int y = y0 + wave;            // this wave's output row
  float* __restrict__ outrow =
      out + ((size_t)(n * (CHANS * ORI) + c * ORI + nidx) * HW + y) * HW;

  for (int xt = 0; xt < NXT; ++xt) {
    const int x0 = xt * 16;
    // per-lane A base: pixel m = nidx, K-pair offset 2*half; ky/kxb folded as
    // compile-time immediates on the unrolled loop.
    const float* __restrict__ abase = &lds[wave * LDSW + x0 + nidx + 2 * half];

    v8f acc = {};
    #pragma unroll
    for (int g = 0; g < 14; ++g) {
      const int off = (g >> 1) * LDSW + (g & 1) * 4;  // ky*LDSW + kxbase
      const float a0 = abase[off];
      const float a1 = abase[off + 1];
      v2f a = {a0, a1};
      acc = __builtin_amdgcn_wmma_f32_16x16x4_f32(
          /*neg_a=*/false, a, /*neg_b=*/false, bmat[g],
          /*c_mod=*/(short)0, acc, /*reuse_a=*/false, /*reuse_b=*/false);
    }

    // D: lane(L<16) -> orientation L, pixels x0+0..7 ; lane(L>=16) -> x0+8..15
    if (nidx < ORI) {
      float* o = outrow + x0 + half * 8;
      v4f lo = {acc[0], acc[1], acc[2], acc[3]};
      v4f hi = {acc[4], acc[5], acc[6], acc[7]};
      *(v4f*)(o)     = lo;
      *(v4f*)(o + 4) = hi;
    }
  }
}

extern "C" void kernel_launch(void* const* d_in, const int* in_sizes, int n_in,
                              void* d_out, int out_size, void* d_ws, size_t ws_size,
                              hipStream_t stream) {
  (void)in_sizes; (void)n_in; (void)d_ws; (void)ws_size; (void)out_size;
  const float* x    = (const float*)d_in[0];
  const float* filt = (const float*)d_in[1];
  float* out        = (float*)d_out;

  const int blocks = BATCH * CHANS * NYB;   // 16*32*28 = 14336
  GaborFilter_wmma_kernel<<<blocks, 256, 0, stream>>>(x, filt, out);
}